// PositionAttention_74285754352198
// MI455X (gfx1250) — compile-verified
//
#include <hip/hip_runtime.h>
#include <hip/hip_bf16.h>
#include <math.h>

typedef __attribute__((ext_vector_type(2))) float v2f;
typedef __attribute__((ext_vector_type(8))) float v8f;

#define PA_MIN_SIGMA 0.05f
#define PA_B 128
#define PA_L 16384
#define PA_D 512
#define PA_H 32

// ---------------------------------------------------------------------------
// Phase 1: fused MLP heads via V_WMMA_F32_16X16X4_F32.
// One wave (32 threads) per 16 batches. GEMM1: (16 x 520pad) @ (520pad x 32)
// for both heads (4 accumulators). GEMM2 (K=32, N=8) scalar from LDS.
// ---------------------------------------------------------------------------
__global__ __launch_bounds__(32) void pa_mlp_kernel(
    const float* __restrict__ dec,        // (128, 512)
    const float* __restrict__ mu_old,     // (128)
    const float* __restrict__ sigma_old,  // (128)
    const float* __restrict__ mc_old,     // (128)
    const float* __restrict__ ma_old,     // (128)
    const float* __restrict__ Wmu1, const float* __restrict__ bmu1,
    const float* __restrict__ Wmu2, const float* __restrict__ bmu2,
    const float* __restrict__ Wsg1, const float* __restrict__ bsg1,
    const float* __restrict__ Wsg2, const float* __restrict__ bsg2,
    const float* __restrict__ conf_temp, const float* __restrict__ conf_bias,
    const int*   __restrict__ source_lengths,
    const int*   __restrict__ step,
    float* __restrict__ out_conf, float* __restrict__ out_mu,
    float* __restrict__ out_sigma)
{
  const int lane = threadIdx.x & 31;
  const int half = lane >> 4;       // 0: K pair {k0,k0+1}; 1: {k0+2,k0+3}
  const int r    = lane & 15;       // row within tile / column within B frag
  const int b    = blockIdx.x * 16 + r;

  const float len     = (float)source_lengths[b];
  const float inv_len = 1.0f / len;
  const float rel_dec = (float)(*step + 1) * inv_len;  // rel_counter[step]*L/len
  const float sh0 = mu_old[b];
  const float sh1 = sigma_old[b] * inv_len;            // relative sigma
  const float sh2 = mc_old[b];
  const float sh3 = ma_old[b];

  v8f accMu0 = {}; v8f accMu1 = {}; v8f accSg0 = {}; v8f accSg1 = {};

  const float* decb = dec + b * PA_D;

  // Main K loop over decoder features (K = 0..511, step 4)
  for (int k0 = 0; k0 < PA_D; k0 += 4) {
    const int kk = k0 + 2 * half;
    v2f a;
    a.x = decb[kk];
    a.y = decb[kk + 1];
    v2f bm0, bm1, bs0, bs1;
    bm0.x = Wmu1[kk * PA_H + r];           bm0.y = Wmu1[(kk + 1) * PA_H + r];
    bm1.x = Wmu1[kk * PA_H + r + 16];      bm1.y = Wmu1[(kk + 1) * PA_H + r + 16];
    bs0.x = Wsg1[kk * PA_H + r];           bs0.y = Wsg1[(kk + 1) * PA_H + r];
    bs1.x = Wsg1[kk * PA_H + r + 16];      bs1.y = Wsg1[(kk + 1) * PA_H + r + 16];
    accMu0 = __builtin_amdgcn_wmma_f32_16x16x4_f32(false, a, false, bm0, (short)0, accMu0, false, false);
    accMu1 = __builtin_amdgcn_wmma_f32_16x16x4_f32(false, a, false, bm1, (short)0, accMu1, false, false);
    accSg0 = __builtin_amdgcn_wmma_f32_16x16x4_f32(false, a, false, bs0, (short)0, accSg0, false, false);
    accSg1 = __builtin_amdgcn_wmma_f32_16x16x4_f32(false, a, false, bs1, (short)0, accSg1, false, false);
  }

  // Tail K = 512..515: the first four "shared" features (all weight rows valid)
  {
    const int kk = 512 + 2 * half;
    v2f a;
    a.x = half ? sh2 : sh0;
    a.y = half ? sh3 : sh1;
    v2f bm0, bm1, bs0, bs1;
    bm0.x = Wmu1[kk * PA_H + r];           bm0.y = Wmu1[(kk + 1) * PA_H + r];
    bm1.x = Wmu1[kk * PA_H + r + 16];      bm1.y = Wmu1[(kk + 1) * PA_H + r + 16];
    bs0.x = Wsg1[kk * PA_H + r];           bs0.y = Wsg1[(kk + 1) * PA_H + r];
    bs1.x = Wsg1[kk * PA_H + r + 16];      bs1.y = Wsg1[(kk + 1) * PA_H + r + 16];
    accMu0 = __builtin_amdgcn_wmma_f32_16x16x4_f32(false, a, false, bm0, (short)0, accMu0, false, false);
    accMu1 = __builtin_amdgcn_wmma_f32_16x16x4_f32(false, a, false, bm1, (short)0, accMu1, false, false);
    accSg0 = __builtin_amdgcn_wmma_f32_16x16x4_f32(false, a, false, bs0, (short)0, accSg0, false, false);
    accSg1 = __builtin_amdgcn_wmma_f32_16x16x4_f32(false, a, false, bs1, (short)0, accSg1, false, false);
  }
  // Tail K = 516..519: rel_dec then zero padding (K=517..519 -> 0)
  {
    v2f a;
    a.x = half ? 0.0f : rel_dec;
    a.y = 0.0f;
    v2f bm0 = {}, bm1 = {}, bs0 = {}, bs1 = {};
    if (!half) {  // only row 516 exists; masked lanes just build zero regs
      bm0.x = Wmu1[516 * PA_H + r];
      bm1.x = Wmu1[516 * PA_H + r + 16];
      bs0.x = Wsg1[516 * PA_H + r];
      bs1.x = Wsg1[516 * PA_H + r + 16];
    }
    accMu0 = __builtin_amdgcn_wmma_f32_16x16x4_f32(false, a, false, bm0, (short)0, accMu0, false, false);
    accMu1 = __builtin_amdgcn_wmma_f32_16x16x4_f32(false, a, false, bm1, (short)0, accMu1, false, false);
    accSg0 = __builtin_amdgcn_wmma_f32_16x16x4_f32(false, a, false, bs0, (short)0, accSg0, false, false);
    accSg1 = __builtin_amdgcn_wmma_f32_16x16x4_f32(false, a, false, bs1, (short)0, accSg1, false, false);
  }

  // Bias + ReLU, spill hidden activations (16x32 per head) to LDS.
  __shared__ float Hmu[16][PA_H + 1];
  __shared__ float Hsg[16][PA_H + 1];
  {
    const float b1m0 = bmu1[r], b1m1 = bmu1[r + 16];
    const float b1s0 = bsg1[r], b1s1 = bsg1[r + 16];
#pragma unroll
    for (int i = 0; i < 8; ++i) {
      const int row = i + 8 * half;      // D layout: VGPR i <-> M = i + 8*(lane/16)
      Hmu[row][r]      = fmaxf(accMu0[i] + b1m0, 0.0f);
      Hmu[row][r + 16] = fmaxf(accMu1[i] + b1m1, 0.0f);
      Hsg[row][r]      = fmaxf(accSg0[i] + b1s0, 0.0f);
      Hsg[row][r + 16] = fmaxf(accSg1[i] + b1s1, 0.0f);
    }
  }
  __syncthreads();

  // GEMM2 (K=32, N=8) + building-block dot, scalar; lanes 0..15, one batch each.
  if (lane < 16) {
    float muw[8], sgw[8];
#pragma unroll
    for (int n = 0; n < 8; ++n) { muw[n] = bmu2[n]; sgw[n] = bsg2[n]; }
#pragma unroll 4
    for (int k = 0; k < PA_H; ++k) {
      const float hm = Hmu[r][k];
      const float hs = Hsg[r][k];
#pragma unroll
      for (int n = 0; n < 8; ++n) {
        muw[n] = fmaf(hm, Wmu2[k * 8 + n], muw[n]);
        sgw[n] = fmaf(hs, Wsg2[k * 8 + n], sgw[n]);
      }
    }
    const float blocks[8] = {sh0, sh1, sh2, sh3, rel_dec, inv_len, -inv_len, 1.0f};
    float mu = 0.0f, sg = 0.0f;
#pragma unroll
    for (int n = 0; n < 8; ++n) {
      mu = fmaf(muw[n], blocks[n], mu);
      sg = fmaf(sgw[n], blocks[n], sg);
    }
    const float sigma = fmaxf(sg, 0.0f) + PA_MIN_SIGMA;
    const float x = (-sigma + fmaxf(conf_bias[0], 0.0f)) / conf_temp[0];
    const float conf = 1.0f / (1.0f + __expf(-x));
    out_conf[b]  = conf;
    out_mu[b]    = mu;
    out_sigma[b] = sigma;
  }
}

// ---------------------------------------------------------------------------
// Phase 2: masked Gaussian attention row, L1-normalized. One block per batch.
// Recompute exp on the write pass instead of re-reading 8 MB from HBM.
// ---------------------------------------------------------------------------
__global__ __launch_bounds__(256) void pa_attn_kernel(
    const int*   __restrict__ source_lengths,
    const float* __restrict__ mu_arr,
    const float* __restrict__ sigma_arr,
    float* __restrict__ pos_attn)
{
  const int b   = blockIdx.x;
  const int len = source_lengths[b];
  const float inv_len   = 1.0f / (float)len;
  const float mu        = mu_arr[b];
  const float inv_sigma = 1.0f / sigma_arr[b];
  float* __restrict__ outb = pos_attn + (size_t)b * PA_L;

  // Pass 1: partial sums of z = exp(-((p - mu)/sigma)^2) over valid region.
  float local = 0.0f;
  for (int i = threadIdx.x; i < len; i += blockDim.x) {
    const float p = (float)(i + 1) * inv_len;
    const float d = (p - mu) * inv_sigma;
    local += __expf(-d * d);
  }
#pragma unroll
  for (int off = 16; off > 0; off >>= 1)
    local += __shfl_down(local, off, 32);

  __shared__ float wsum[8];
  const int wid = threadIdx.x >> 5;
  if ((threadIdx.x & 31) == 0) wsum[wid] = local;
  __syncthreads();
  float total = 0.0f;
#pragma unroll
  for (int w = 0; w < 8; ++w) total += wsum[w];
  const float scale = 1.0f / fmaxf(total, 1e-12f);

  // Pass 2: write normalized row (zeros beyond source length).
  for (int i = threadIdx.x; i < PA_L; i += blockDim.x) {
    float z = 0.0f;
    if (i < len) {
      const float p = (float)(i + 1) * inv_len;
      const float d = (p - mu) * inv_sigma;
      z = __expf(-d * d) * scale;
    }
    outb[i] = z;
  }
}

extern "C" void kernel_launch(void* const* d_in, const int* in_sizes, int n_in,
                              void* d_out, int out_size, void* d_ws, size_t ws_size,
                              hipStream_t stream) {
  const float* dec        = (const float*)d_in[0];
  // d_in[1] = encoder_outputs: values are NEVER used by the reference (shape only).
  const float* mu_old     = (const float*)d_in[2];
  const float* sigma_old  = (const float*)d_in[3];
  const float* mc_old     = (const float*)d_in[4];
  const float* ma_old     = (const float*)d_in[5];
  const float* Wmu1       = (const float*)d_in[6];
  const float* bmu1       = (const float*)d_in[7];
  const float* Wmu2       = (const float*)d_in[8];
  const float* bmu2       = (const float*)d_in[9];
  const float* Wsg1       = (const float*)d_in[10];
  const float* bsg1       = (const float*)d_in[11];
  const float* Wsg2       = (const float*)d_in[12];
  const float* bsg2       = (const float*)d_in[13];
  const float* conf_temp  = (const float*)d_in[14];
  const float* conf_bias  = (const float*)d_in[15];
  const int*   lens       = (const int*)d_in[16];
  const int*   step       = (const int*)d_in[17];

  float* out       = (float*)d_out;
  float* pos_attn  = out;                               // (B,1,L)
  float* out_conf  = out + (size_t)PA_B * PA_L;         // (B,1,1)
  float* out_mu    = out_conf + PA_B;                   // (B,1,1)
  float* out_sigma = out_mu + PA_B;                     // (B,1,1)

  pa_mlp_kernel<<<PA_B / 16, 32, 0, stream>>>(
      dec, mu_old, sigma_old, mc_old, ma_old,
      Wmu1, bmu1, Wmu2, bmu2, Wsg1, bsg1, Wsg2, bsg2,
      conf_temp, conf_bias, lens, step,
      out_conf, out_mu, out_sigma);

  pa_attn_kernel<<<PA_B, 256, 0, stream>>>(lens, out_mu, out_sigma, pos_attn);
}